// GraphRecommender_36077725286617
// MI455X (gfx1250) — compile-verified
//
#include <hip/hip_runtime.h>
#include <math.h>

typedef __attribute__((ext_vector_type(2))) float v2f;
typedef __attribute__((ext_vector_type(8))) float v8f;

#define NNODES 40000
#define DEG    12
#define BATCH  1024
#define SEQL   50
#define DIM    256
#define DIM2   512
#define NROWS  (BATCH * SEQL)   // 51200

#define WMMA_F32(a, b, c) \
  __builtin_amdgcn_wmma_f32_16x16x4_f32(false, (a), false, (b), (short)0, (c), false, false)

// ---------------------------------------------------------------------------
// K1: graph conv  ge[n] = 0.5*(emb[n] + sum_j vals[12n+j]*emb[cols[12n+j]])
// one node per 64-thread block (2 waves), float4 per thread
// ---------------------------------------------------------------------------
__global__ void k_graphconv(const float* __restrict__ emb,
                            const int* __restrict__ cols,
                            const float* __restrict__ vals,
                            float* __restrict__ ge) {
  const int n = blockIdx.x;
  const int t = threadIdx.x;           // 0..63 -> 4 channels each
  const int ebase = n * DEG;
  float4 acc = make_float4(0.f, 0.f, 0.f, 0.f);
#pragma unroll
  for (int j = 0; j < DEG; ++j) {
    const int c = cols[ebase + j];
    const float v = vals[ebase + j];
    const float4 x = *(const float4*)(emb + (size_t)c * DIM + t * 4);
    acc.x += v * x.x; acc.y += v * x.y; acc.z += v * x.z; acc.w += v * x.w;
  }
  const float4 s = *(const float4*)(emb + (size_t)n * DIM + t * 4);
  float4 o;
  o.x = 0.5f * (s.x + acc.x); o.y = 0.5f * (s.y + acc.y);
  o.z = 0.5f * (s.z + acc.z); o.w = 0.5f * (s.w + acc.w);
  *(float4*)(ge + (size_t)n * DIM + t * 4) = o;
}

// ---------------------------------------------------------------------------
// K2: tgt[1024x256] = TE[1024x512] @ wf_w[256x512]^T + wf_b
// fp32 WMMA 16x16x4. 4 waves/block, wave -> one 16x16 N-tile.
// A frag: lane l holds row l%16; v0=K(2*(l/16)), v1=K(2*(l/16)+1) -> float2
// B frag: lane l holds col l%16 of B (= row l%16 of wf_w), same K split.
// ---------------------------------------------------------------------------
__global__ void k_target_proj(const float* __restrict__ te,
                              const float* __restrict__ wf_w,
                              const float* __restrict__ wf_b,
                              float* __restrict__ tgt) {
  const int lane = threadIdx.x & 31;
  const int wave = threadIdx.x >> 5;
  const int mt = blockIdx.x;                 // 0..63
  const int nt = blockIdx.y * 4 + wave;      // 0..15
  const int ml = lane & 15;
  const int g  = lane >> 4;
  const float* arow = te   + (size_t)(mt * 16 + ml) * DIM2 + 2 * g;
  const float* brow = wf_w + (size_t)(nt * 16 + ml) * DIM2 + 2 * g;
  v8f c = {};
  for (int kb = 0; kb < DIM2; kb += 4) {
    v2f a = *(const v2f*)(arow + kb);
    v2f b = *(const v2f*)(brow + kb);
    c = WMMA_F32(a, b, c);
  }
  const int ncol = nt * 16 + ml;
  const float bias = wf_b[ncol];
#pragma unroll
  for (int v = 0; v < 8; ++v) {
    const int m = mt * 16 + g * 8 + v;       // C/D: lane half g -> rows 8g..8g+7
    tgt[(size_t)m * DIM + ncol] = c[v] + bias;
  }
}

// ---------------------------------------------------------------------------
// K3a: tproj[1024x256] = tgt @ at_w2[256x256] + at_bias   (B is K-major)
// ---------------------------------------------------------------------------
__global__ void k_tproj(const float* __restrict__ tgt,
                        const float* __restrict__ at_w2,
                        const float* __restrict__ at_bias,
                        float* __restrict__ tproj) {
  const int lane = threadIdx.x & 31;
  const int wave = threadIdx.x >> 5;
  const int mt = blockIdx.x;
  const int nt = blockIdx.y * 4 + wave;
  const int ml = lane & 15;
  const int g  = lane >> 4;
  const int ncol = nt * 16 + ml;
  const float* arow = tgt + (size_t)(mt * 16 + ml) * DIM + 2 * g;
  v8f c = {};
  for (int kb = 0; kb < DIM; kb += 4) {
    v2f a = *(const v2f*)(arow + kb);
    v2f b;
    b.x = at_w2[(size_t)(kb + 2 * g)     * DIM + ncol];
    b.y = at_w2[(size_t)(kb + 2 * g + 1) * DIM + ncol];
    c = WMMA_F32(a, b, c);
  }
  const float bias = at_bias[ncol];
#pragma unroll
  for (int v = 0; v < 8; ++v) {
    const int m = mt * 16 + g * 8 + v;
    tproj[(size_t)m * DIM + ncol] = c[v] + bias;
  }
}

// ---------------------------------------------------------------------------
// K3b: alpha[b] = sigmoid(dot(tgt[b], alphaw_w) + alphaw_b) + 1 (==1 -> 1.00001)
// one wave per row
// ---------------------------------------------------------------------------
__global__ void k_alpha(const float* __restrict__ tgt,
                        const float* __restrict__ aw,
                        const float* __restrict__ ab,
                        float* __restrict__ alpha) {
  const int lane = threadIdx.x & 31;
  const int wave = threadIdx.x >> 5;
  const int b = blockIdx.x * 8 + wave;
  const float* row = tgt + (size_t)b * DIM;
  float s = 0.f;
#pragma unroll
  for (int i = 0; i < 8; ++i) s += row[lane + 32 * i] * aw[lane + 32 * i];
  for (int off = 16; off; off >>= 1) s += __shfl_xor(s, off, 32);
  if (lane == 0) {
    const float x = s + ab[0];
    float a = 1.0f / (1.0f + expf(-x)) + 1.0f;
    if (a == 1.0f) a = 1.00001f;
    alpha[b] = a;
  }
}

// ---------------------------------------------------------------------------
// K4: scores[m] = sum_n relu( (ge[items[m]] @ at_w1)[n] + tproj[m/50, n] ) * at_w0[n]
// block = 256 thr (8 waves). Block handles 64 gathered rows (4 M-tiles) so each
// B fragment is reused by 4 A fragments (4x less at_w1 L2 traffic).
// wave w -> N-tiles 2w, 2w+1 (cols 32w..32w+31). 8 WMMAs per wave per K-step.
// ---------------------------------------------------------------------------
__global__ void k_scores(const float* __restrict__ ge,
                         const int* __restrict__ items,
                         const float* __restrict__ at_w1,
                         const float* __restrict__ tproj,
                         const float* __restrict__ at_w0,
                         float* __restrict__ scores) {
  __shared__ float srow[64];
  const int lane = threadIdx.x & 31;
  const int wave = threadIdx.x >> 5;
  const int blk = blockIdx.x;                // 0..799, 64 rows each
  const int ml = lane & 15;
  const int g  = lane >> 4;
  if (threadIdx.x < 64) srow[threadIdx.x] = 0.0f;
  __syncthreads();

  const int ncol0 = (wave * 2) * 16 + ml;
  const int ncol1 = ncol0 + 16;

  const float* arow[4];
#pragma unroll
  for (int s = 0; s < 4; ++s) {
    const int p = blk * 64 + s * 16 + ml;
    arow[s] = ge + (size_t)items[p] * DIM + 2 * g;
    __builtin_prefetch(arow[s], 0, 3);       // global_prefetch_b8: warm L0 for gathered rows
  }

  v8f c00 = {}, c01 = {}, c10 = {}, c11 = {};
  v8f c20 = {}, c21 = {}, c30 = {}, c31 = {};
#pragma unroll 4
  for (int kb = 0; kb < DIM; kb += 4) {
    const size_t r0 = (size_t)(kb + 2 * g) * DIM;
    const size_t r1 = r0 + DIM;
    v2f b0, b1;
    b0.x = at_w1[r0 + ncol0]; b0.y = at_w1[r1 + ncol0];
    b1.x = at_w1[r0 + ncol1]; b1.y = at_w1[r1 + ncol1];
    v2f a0 = *(const v2f*)(arow[0] + kb);
    v2f a1 = *(const v2f*)(arow[1] + kb);
    v2f a2 = *(const v2f*)(arow[2] + kb);
    v2f a3 = *(const v2f*)(arow[3] + kb);
    c00 = WMMA_F32(a0, b0, c00); c01 = WMMA_F32(a0, b1, c01);
    c10 = WMMA_F32(a1, b0, c10); c11 = WMMA_F32(a1, b1, c11);
    c20 = WMMA_F32(a2, b0, c20); c21 = WMMA_F32(a2, b1, c21);
    c30 = WMMA_F32(a3, b0, c30); c31 = WMMA_F32(a3, b1, c31);
  }

  const float w00 = at_w0[ncol0];
  const float w01 = at_w0[ncol1];
  auto emit = [&](int s, const v8f& cA, const v8f& cB) {
#pragma unroll
    for (int v = 0; v < 8; ++v) {
      const int m = blk * 64 + s * 16 + g * 8 + v;
      const size_t tb = (size_t)(m / SEQL) * DIM;
      float x0 = cA[v] + tproj[tb + ncol0];
      float x1 = cB[v] + tproj[tb + ncol1];
      x0 = x0 > 0.f ? x0 : 0.f;
      x1 = x1 > 0.f ? x1 : 0.f;
      float p = x0 * w00 + x1 * w01;
      p += __shfl_xor(p, 1, 32);
      p += __shfl_xor(p, 2, 32);
      p += __shfl_xor(p, 4, 32);
      p += __shfl_xor(p, 8, 32);
      if (ml == 0) atomicAdd(&srow[s * 16 + g * 8 + v], p);
    }
  };
  emit(0, c00, c01); emit(1, c10, c11);
  emit(2, c20, c21); emit(3, c30, c31);
  __syncthreads();
  if (threadIdx.x < 64) scores[blk * 64 + threadIdx.x] = srow[threadIdx.x];
}

// ---------------------------------------------------------------------------
// K5: entmax-bisect over L=50 per batch row; one wave per row.
// lane handles elements {lane, lane+32}
// ---------------------------------------------------------------------------
__global__ void k_entmax(const float* __restrict__ scores,
                         const float* __restrict__ alpha,
                         float* __restrict__ attn) {
  const int lane = threadIdx.x & 31;
  const int wave = threadIdx.x >> 5;
  const int b = blockIdx.x * 8 + wave;       // 1024 rows
  const float al  = alpha[b];
  const float am1 = al - 1.0f;
  const float inv = 1.0f / am1;
  const float* srow = scores + b * SEQL;
  const int i1 = lane + 32;
  const bool v1 = (i1 < SEQL);
  const float xa0 = srow[lane] * am1;
  const float xa1 = v1 ? srow[i1] * am1 : -INFINITY;

  float mx = fmaxf(xa0, xa1);
  for (int off = 16; off; off >>= 1) mx = fmaxf(mx, __shfl_xor(mx, off, 32));

  float tau_lo = mx - 1.0f;
  const float tau_hi = mx - powf(1.0f / (float)SEQL, am1);

  float s = (xa0 - tau_lo > 0.f ? powf(xa0 - tau_lo, inv) : 0.f)
          + (xa1 - tau_lo > 0.f ? powf(xa1 - tau_lo, inv) : 0.f);
  for (int off = 16; off; off >>= 1) s += __shfl_xor(s, off, 32);
  const float f_lo = s - 1.0f;

  float dm = tau_hi - tau_lo;
  float tau_m = tau_lo;
  for (int it = 0; it < 50; ++it) {
    dm *= 0.5f;
    tau_m = tau_lo + dm;
    float sm = (xa0 - tau_m > 0.f ? powf(xa0 - tau_m, inv) : 0.f)
             + (xa1 - tau_m > 0.f ? powf(xa1 - tau_m, inv) : 0.f);
    for (int off = 16; off; off >>= 1) sm += __shfl_xor(sm, off, 32);
    if ((sm - 1.0f) * f_lo >= 0.f) tau_lo = tau_m;   // uniform across wave
  }
  const float p0 = (xa0 - tau_m > 0.f ? powf(xa0 - tau_m, inv) : 0.f);
  const float p1 = (xa1 - tau_m > 0.f ? powf(xa1 - tau_m, inv) : 0.f);
  float ps = p0 + p1;
  for (int off = 16; off; off >>= 1) ps += __shfl_xor(ps, off, 32);
  attn[b * SEQL + lane] = p0 / ps;
  if (v1) attn[b * SEQL + i1] = p1 / ps;
}

// ---------------------------------------------------------------------------
// K6: c[b,:] = selu( sum_l attn[b,l]*ge[items[b,l]] ); out = c / ||c||2
// one block (256 thr) per batch row
// ---------------------------------------------------------------------------
__global__ void k_final(const float* __restrict__ ge,
                        const int* __restrict__ items,
                        const float* __restrict__ attn,
                        float* __restrict__ out) {
  __shared__ int   sitem[SEQL];
  __shared__ float sattn[SEQL];
  __shared__ float red[8];
  const int b = blockIdx.x;
  const int d = threadIdx.x;
  if (d < SEQL) { sitem[d] = items[b * SEQL + d]; sattn[d] = attn[b * SEQL + d]; }
  __syncthreads();
  float acc = 0.f;
  for (int l = 0; l < SEQL; ++l)
    acc += sattn[l] * ge[(size_t)sitem[l] * DIM + d];
  const float kScale = 1.0507009873554804934193349852946f;
  const float kAlpha = 1.6732632423543772848170429916717f;
  const float c = acc > 0.f ? kScale * acc : kScale * kAlpha * (expf(acc) - 1.0f);
  float ss = c * c;
  for (int off = 16; off; off >>= 1) ss += __shfl_xor(ss, off, 32);
  if ((threadIdx.x & 31) == 0) red[threadIdx.x >> 5] = ss;
  __syncthreads();
  if (threadIdx.x == 0) {
    float t = 0.f;
#pragma unroll
    for (int i = 0; i < 8; ++i) t += red[i];
    red[0] = t;
  }
  __syncthreads();
  out[(size_t)b * DIM + d] = c / sqrtf(red[0]);
}

// ---------------------------------------------------------------------------
extern "C" void kernel_launch(void* const* d_in, const int* in_sizes, int n_in,
                              void* d_out, int out_size, void* d_ws, size_t ws_size,
                              hipStream_t stream) {
  (void)in_sizes; (void)n_in; (void)out_size; (void)ws_size;
  const int*   items    = (const int*)d_in[0];
  // d_in[1] inputs_seq, d_in[2] alias_inputs: unused by the reference math
  const float* te       = (const float*)d_in[3];
  const float* item_emb = (const float*)d_in[4];
  // d_in[5] adj_rows == repeat(arange(N),12) by construction -> implicit
  const int*   adj_cols = (const int*)d_in[6];
  const float* adj_vals = (const float*)d_in[7];
  const float* wf_w     = (const float*)d_in[8];
  const float* wf_b     = (const float*)d_in[9];
  const float* alphaw_w = (const float*)d_in[10];
  const float* alphaw_b = (const float*)d_in[11];
  const float* at_w0    = (const float*)d_in[12];
  const float* at_w1    = (const float*)d_in[13];
  const float* at_w2    = (const float*)d_in[14];
  const float* at_bias  = (const float*)d_in[15];
  float* out = (float*)d_out;

  float* ws     = (float*)d_ws;
  float* ge     = ws;                                   // 40000*256
  float* tgt    = ge     + (size_t)NNODES * DIM;        // 1024*256
  float* tproj  = tgt    + (size_t)BATCH * DIM;         // 1024*256
  float* alpha  = tproj  + (size_t)BATCH * DIM;         // 1024
  float* scores = alpha  + BATCH;                       // 51200
  float* attn   = scores + NROWS;                       // 51200

  k_graphconv<<<NNODES, 64, 0, stream>>>(item_emb, adj_cols, adj_vals, ge);
  k_target_proj<<<dim3(BATCH / 16, 4), 128, 0, stream>>>(te, wf_w, wf_b, tgt);
  k_tproj<<<dim3(BATCH / 16, 4), 128, 0, stream>>>(tgt, at_w2, at_bias, tproj);
  k_alpha<<<BATCH / 8, 256, 0, stream>>>(tgt, alphaw_w, alphaw_b, alpha);
  k_scores<<<NROWS / 64, 256, 0, stream>>>(ge, items, at_w1, tproj, at_w0, scores);
  k_entmax<<<BATCH / 8, 256, 0, stream>>>(scores, alpha, attn);
  k_final<<<BATCH, 256, 0, stream>>>(ge, items, attn, out);
}